// SurvRankingLoss_27032524161730
// MI455X (gfx1250) — compile-verified
//
#include <hip/hip_runtime.h>

typedef __attribute__((ext_vector_type(2))) float v2f;
typedef __attribute__((ext_vector_type(8))) float v8f;

#define B_N     8192
#define I_BAND  128    // rows of i per block = 8 waves * 16
#define J_CHUNK 2048   // cols of j per block
#define THREADS 256

// Native CDNA5 32-bit tanh (TRANS pipe). Prefer the clang builtin; fall back
// to inline asm with a v_nop (ISA 7.4: TRANS result needs 1 independent op
// before use; compiler can't schedule around opaque asm).
__device__ __forceinline__ float tanh_f32(float x) {
#if __has_builtin(__builtin_amdgcn_tanhf)
    return __builtin_amdgcn_tanhf(x);
#elif __has_builtin(__builtin_amdgcn_tanh_f32)
    return __builtin_amdgcn_tanh_f32(x);
#else
    float y;
    asm volatile("v_tanh_f32 %0, %1\n\tv_nop\n\tv_nop" : "=v"(y) : "v"(x));
    return y;
#endif
}

// Each wave computes 16x16 half-difference tiles with V_WMMA_F32_16X16X4_F32
// (rank-2 update: D = (0.5*z_i)*1^T + 1*(-0.5*z_j)^T = r/2), then accumulates
// masked tanh(r/2) and the pair count. sigmoid is reconstructed at the end:
//   total = 0.5*n_pairs + 0.5*sum(tanh(r/2))  =>  1 trans op per pair.
__global__ __launch_bounds__(THREADS)
void surv_pair_kernel(const float* __restrict__ z,
                      const float* __restrict__ c_t,
                      float* __restrict__ psum,
                      unsigned int* __restrict__ pcnt)
{
    __shared__ float s_zj[J_CHUNK];
    __shared__ float s_tj[J_CHUNK];
    __shared__ float s_wsum[THREADS / 32];
    __shared__ unsigned int s_wcnt[THREADS / 32];

    const int tid  = threadIdx.x;
    const int wave = tid >> 5;
    const int lane = tid & 31;
    const int n    = lane & 15;   // column within 16x16 tile
    const int hi   = lane >> 4;   // selects K=2,3 half (A/B) and M+8 half (D)

    const int i_base = blockIdx.x * I_BAND;
    const int j_base = blockIdx.y * J_CHUNK;

    // Stage this block's j-chunk (z_j, t_j) into LDS.
    for (int k = tid; k < J_CHUNK; k += THREADS) {
        const int j = j_base + k;
        s_zj[k] = z[j];
        s_tj[k] = c_t[2 * j + 1];
    }
    __syncthreads();

    // A operand (16x4 f32), loop-invariant over j:
    // lanes 0-15 rows M=lane: VGPR0 = K0 = 0.5*z_i, VGPR1 = K1 = 1.0
    // lanes 16-31 hold K=2,3 (unused -> 0).
    const int i0 = i_base + wave * 16;
    v2f a;
    a.x = (hi == 0) ? 0.5f * z[i0 + n] : 0.0f;
    a.y = (hi == 0) ? 1.0f             : 0.0f;

    // Per-lane row metadata: this lane's D rows are m = v + 8*hi, v=0..7.
    // teff = +inf when censored -> "times[i] < times[j]" never holds.
    float teff[8];
#pragma unroll
    for (int v = 0; v < 8; ++v) {
        const int i     = i0 + v + 8 * hi;
        const float cen = c_t[2 * i + 0];
        const float t   = c_t[2 * i + 1];
        teff[v] = ((1.0f - cen) > 0.0f) ? t : 3.0e38f;
    }

    float sum = 0.0f;          // sum of masked tanh(r/2)
    unsigned int cnt = 0u;     // exact pair count

#pragma unroll 2
    for (int jt = 0; jt < J_CHUNK; jt += 16) {
        const float zj = s_zj[jt + n];
        const float tj = s_tj[jt + n];

        // B operand (4x16 f32): lanes 0-15: VGPR0 = K0 row = 1, VGPR1 = K1 row = -0.5*z_j
        v2f b;
        b.x = (hi == 0) ? 1.0f        : 0.0f;
        b.y = (hi == 0) ? -0.5f * zj  : 0.0f;

        v8f c = {};
        // D[m][n] = 0.5*(z_i[m] - z_j[n])   (exact f32 rank-2 update)
        v8f d = __builtin_amdgcn_wmma_f32_16x16x4_f32(
            false, a, false, b, (short)0, c, false, false);

        // Branchless masked accumulation: tanh computed unconditionally,
        // addend selected via cndmask -> no EXEC churn, full pipelining.
#pragma unroll
        for (int v = 0; v < 8; ++v) {
            const float th = tanh_f32(d[v]);
            const bool  p  = teff[v] < tj;
            sum += p ? th : 0.0f;
            cnt += p ? 1u : 0u;
        }
    }

    // Deterministic fixed-order wave32 tree reduction.
#pragma unroll
    for (int off = 16; off > 0; off >>= 1) {
        sum += __shfl_down(sum, off, 32);
        cnt += __shfl_down(cnt, off, 32);
    }
    if (lane == 0) { s_wsum[wave] = sum; s_wcnt[wave] = cnt; }
    __syncthreads();

    if (tid == 0) {
        float bs = 0.0f; unsigned int bc = 0u;
#pragma unroll
        for (int w = 0; w < THREADS / 32; ++w) { bs += s_wsum[w]; bc += s_wcnt[w]; }
        const int p = blockIdx.y * gridDim.x + blockIdx.x;
        psum[p] = bs;
        pcnt[p] = bc;
    }
}

// Single-wave deterministic finalize:
//   total = 0.5*n + 0.5*sum_tanh  =>  loss = -(0.5*sum_tanh/n + 0.5)
__global__ void surv_finalize_kernel(const float* __restrict__ psum,
                                     const unsigned int* __restrict__ pcnt,
                                     float* __restrict__ out, int nparts)
{
    const int lane = threadIdx.x;
    float s = 0.0f; unsigned int c = 0u;
    for (int k = lane; k < nparts; k += 32) { s += psum[k]; c += pcnt[k]; }
#pragma unroll
    for (int off = 16; off > 0; off >>= 1) {
        s += __shfl_down(s, off, 32);
        c += __shfl_down(c, off, 32);
    }
    if (lane == 0) {
        out[0] = (c > 0u) ? -(0.5f * (s / (float)c) + 0.5f) : -0.0f;
    }
}

extern "C" void kernel_launch(void* const* d_in, const int* in_sizes, int n_in,
                              void* d_out, int out_size, void* d_ws, size_t ws_size,
                              hipStream_t stream) {
    const float* z   = (const float*)d_in[0];   // (B, 1) f32
    const float* c_t = (const float*)d_in[1];   // (B, 2) f32: [censorship, times]
    float* out = (float*)d_out;                 // scalar f32

    const int NPART = (B_N / I_BAND) * (B_N / J_CHUNK);   // 64 * 4 = 256
    float* psum        = (float*)d_ws;
    unsigned int* pcnt = (unsigned int*)((char*)d_ws + NPART * sizeof(float));

    dim3 grid(B_N / I_BAND, B_N / J_CHUNK);
    surv_pair_kernel<<<grid, THREADS, 0, stream>>>(z, c_t, psum, pcnt);
    surv_finalize_kernel<<<1, 32, 0, stream>>>(psum, pcnt, out, NPART);
}